// SparseLinear_5643587027243
// MI455X (gfx1250) — compile-verified
//
#include <hip/hip_runtime.h>

// ---------------------------------------------------------------------------
// SparseLinear (COO SpMM + bias) for MI455X / gfx1250, wave32.
//
// Pipeline (needs ~33MB workspace):
//   1. k_zero       : counts[n] = 0
//   2. k_hist       : counts[row[i]]++                      (u32 atomics)
//   3. k_scan_block : per-1024 exclusive scan of counts -> row_start (partial)
//   4. k_scan_sums  : serial scan of block sums (tiny)
//   5. k_finalize   : row_start final; fill[] = row_start copy
//   6. k_scatter    : pairs[pos] = {col | (row&15)<<16, val}  (bin by row)
//   7. k_spmm_wmma  : 1 wave per 16-row block; chunks of 4 nnz accumulated
//                     via V_WMMA_F32_16X16X4_F32 (indicator-matrix trick),
//                     single non-atomic store of each output row + bias.
// Fallback (workspace too small): bias-init + f32 global atomics.
// ---------------------------------------------------------------------------

typedef float v2f __attribute__((ext_vector_type(2)));
typedef float v8f __attribute__((ext_vector_type(8)));

__global__ void k_zero(unsigned* __restrict__ p, int n) {
  int i = blockIdx.x * blockDim.x + threadIdx.x;
  if (i < n) p[i] = 0u;
}

__global__ void k_hist(const int* __restrict__ rows, unsigned* __restrict__ counts, int nnz) {
  int i = blockIdx.x * blockDim.x + threadIdx.x;
  if (i < nnz) atomicAdd(&counts[rows[i]], 1u);
}

// Exclusive scan of 1024 counts per workgroup (256 thr x 4 elems), Hillis-Steele in LDS.
__global__ __launch_bounds__(256) void k_scan_block(const unsigned* __restrict__ counts,
                                                    unsigned* __restrict__ scanned,
                                                    unsigned* __restrict__ blocksums, int n) {
  __shared__ unsigned lds[256];
  const int t = threadIdx.x;
  const int base = blockIdx.x * 1024 + t * 4;
  unsigned c0 = (base + 0 < n) ? counts[base + 0] : 0u;
  unsigned c1 = (base + 1 < n) ? counts[base + 1] : 0u;
  unsigned c2 = (base + 2 < n) ? counts[base + 2] : 0u;
  unsigned c3 = (base + 3 < n) ? counts[base + 3] : 0u;
  const unsigned s = c0 + c1 + c2 + c3;
  lds[t] = s;
  __syncthreads();
  for (int off = 1; off < 256; off <<= 1) {
    unsigned v = (t >= off) ? lds[t - off] : 0u;
    __syncthreads();
    lds[t] += v;
    __syncthreads();
  }
  if (t == 255) blocksums[blockIdx.x] = lds[255];
  unsigned p = lds[t] - s;  // exclusive prefix
  if (base + 0 < n) scanned[base + 0] = p; p += c0;
  if (base + 1 < n) scanned[base + 1] = p; p += c1;
  if (base + 2 < n) scanned[base + 2] = p; p += c2;
  if (base + 3 < n) scanned[base + 3] = p;
}

__global__ void k_scan_sums(unsigned* __restrict__ blocksums, int numBlocks) {
  if (blockIdx.x == 0 && threadIdx.x == 0) {
    unsigned run = 0;
    for (int b = 0; b < numBlocks; ++b) { unsigned v = blocksums[b]; blocksums[b] = run; run += v; }
  }
}

__global__ void k_finalize(unsigned* __restrict__ row_start, unsigned* __restrict__ fill,
                           const unsigned* __restrict__ blocksums, int n, unsigned total) {
  int i = blockIdx.x * blockDim.x + threadIdx.x;
  if (i < n) {
    unsigned v = row_start[i] + blocksums[i >> 10];
    row_start[i] = v;
    fill[i] = v;
  } else if (i == n) {
    row_start[n] = total;
  }
}

// Bin nnz by row: record = { col | (row&15)<<16 , f32 value bits }.
__global__ void k_scatter(const int* __restrict__ rows, const int* __restrict__ cols,
                          const float* __restrict__ vals, unsigned* __restrict__ fill,
                          uint2* __restrict__ pairs, int nnz) {
  int i = blockIdx.x * blockDim.x + threadIdx.x;
  if (i >= nnz) return;
  int r = rows[i];
  unsigned pos = atomicAdd(&fill[r], 1u);
  uint2 p;
  p.x = ((unsigned)cols[i] & 0xFFFFu) | (((unsigned)r & 15u) << 16);
  p.y = __float_as_uint(vals[i]);
  pairs[pos] = p;
}

// One wave per 16-row block. For each chunk of 4 binned nnz:
//   A[16x4] (f32): A[m][k] = (localrow_k == m) ? v_k : 0   (indicator * value)
//   B[4x64] (f32): B[k][:] = W[c_k][:]                      (gathered, L2-hot)
//   Out[16x64]   += A @ B  via 4x V_WMMA_F32_16X16X4_F32 (bit-exact f32 path).
// ISA 7.12.2 layouts: A 16x4 -> v0:{K0|K2}, v1:{K1|K3} per half-wave;
// B 4x16 -> v0:{K0-row|K2-row}, v1:{K1|K3}; C/D vgpr j -> M=j (lo) / M=j+8 (hi).
__global__ __launch_bounds__(32) void k_spmm_wmma(const uint2* __restrict__ pairs,
                                                  const unsigned* __restrict__ row_start,
                                                  const float* __restrict__ W,
                                                  const float* __restrict__ bias,
                                                  float* __restrict__ out, int n) {
  const int lane = threadIdx.x & 31;
  const int r0 = blockIdx.x << 4;
  int rend = r0 + 16; if (rend > n) rend = n;
  const unsigned segBeg = row_start[r0];
  const unsigned segEnd = row_start[rend];

  const bool hi = lane >= 16;
  const int lrow = lane & 15;

  v8f acc0 = {}, acc1 = {}, acc2 = {}, acc3 = {};

  for (unsigned base = segBeg; base < segEnd; base += 32u) {
    const unsigned my = base + (unsigned)lane;
    uint2 rec; rec.x = 0u; rec.y = 0u;           // pad: col=0, lrow=0, val=0 (harmless)
    if (my < segEnd) rec = pairs[my];
    if (my + 32u < segEnd) __builtin_prefetch(&pairs[my + 32u], 0, 0);
    const unsigned rem = segEnd - base;
    const unsigned cnt = rem < 32u ? rem : 32u;

    for (unsigned s4 = 0; s4 < cnt; s4 += 4u) {
      // Broadcast 4 records (uniform lane index -> v_readlane, lands in SGPRs).
      const int m0 = __builtin_amdgcn_readlane((int)rec.x, (int)s4 + 0);
      const int m1 = __builtin_amdgcn_readlane((int)rec.x, (int)s4 + 1);
      const int m2 = __builtin_amdgcn_readlane((int)rec.x, (int)s4 + 2);
      const int m3 = __builtin_amdgcn_readlane((int)rec.x, (int)s4 + 3);
      const float v0 = __uint_as_float((unsigned)__builtin_amdgcn_readlane((int)rec.y, (int)s4 + 0));
      const float v1 = __uint_as_float((unsigned)__builtin_amdgcn_readlane((int)rec.y, (int)s4 + 1));
      const float v2 = __uint_as_float((unsigned)__builtin_amdgcn_readlane((int)rec.y, (int)s4 + 2));
      const float v3 = __uint_as_float((unsigned)__builtin_amdgcn_readlane((int)rec.y, (int)s4 + 3));

      // Low half-wave carries K=0,1; high half-wave carries K=2,3.
      const int   mA0 = hi ? m2 : m0; const float vA0 = hi ? v2 : v0;
      const int   mA1 = hi ? m3 : m1; const float vA1 = hi ? v3 : v1;

      v2f A;
      A.x = (((mA0 >> 16) & 15) == lrow) ? vA0 : 0.0f;   // A[M=lrow][K0/K2]
      A.y = (((mA1 >> 16) & 15) == lrow) ? vA1 : 0.0f;   // A[M=lrow][K1/K3]

      const float* Wr0 = W + (size_t)(mA0 & 0xFFFF) * 64u;  // row c_{K0/K2}
      const float* Wr1 = W + (size_t)(mA1 & 0xFFFF) * 64u;  // row c_{K1/K3}

      v2f B0, B1, B2, B3;                       // 4 column-slices of the gathered 4x64
      B0.x = Wr0[lrow +  0]; B0.y = Wr1[lrow +  0];
      B1.x = Wr0[lrow + 16]; B1.y = Wr1[lrow + 16];
      B2.x = Wr0[lrow + 32]; B2.y = Wr1[lrow + 32];
      B3.x = Wr0[lrow + 48]; B3.y = Wr1[lrow + 48];

      acc0 = __builtin_amdgcn_wmma_f32_16x16x4_f32(false, A, false, B0, (short)0, acc0, false, false);
      acc1 = __builtin_amdgcn_wmma_f32_16x16x4_f32(false, A, false, B1, (short)0, acc1, false, false);
      acc2 = __builtin_amdgcn_wmma_f32_16x16x4_f32(false, A, false, B2, (short)0, acc2, false, false);
      acc3 = __builtin_amdgcn_wmma_f32_16x16x4_f32(false, A, false, B3, (short)0, acc3, false, false);
    }
  }

  const float b0 = bias[lrow +  0];
  const float b1 = bias[lrow + 16];
  const float b2 = bias[lrow + 32];
  const float b3 = bias[lrow + 48];
#pragma unroll
  for (int j = 0; j < 8; ++j) {
    const int r = r0 + j + (hi ? 8 : 0);
    if (r < n) {
      float* o = out + (size_t)r * 64u + (unsigned)lrow;
      o[ 0] = acc0[j] + b0;
      o[16] = acc1[j] + b1;
      o[32] = acc2[j] + b2;
      o[48] = acc3[j] + b3;
    }
  }
}

// ------------------------- fallback (no workspace) -------------------------
__global__ void k_init_out(float* __restrict__ out, const float* __restrict__ bias, int total) {
  int i = blockIdx.x * blockDim.x + threadIdx.x;
  if (i < total) out[i] = bias[i & 63];
}

__global__ void k_atomic_spmm(const int* __restrict__ rows, const int* __restrict__ cols,
                              const float* __restrict__ vals, const float* __restrict__ W,
                              float* __restrict__ out, int nnz) {
  long long i = (long long)blockIdx.x * blockDim.x + threadIdx.x;
  long long e = i >> 3;
  int g = (int)(i & 7);
  if (e >= (long long)nnz) return;
  int r = rows[e], c = cols[e];
  float v = vals[e];
  const float* Wr = W + (size_t)c * 64u + g * 8;
  float* o = out + (size_t)r * 64u + g * 8;
#pragma unroll
  for (int k = 0; k < 8; ++k) atomicAdd(&o[k], v * Wr[k]);
}

// ---------------------------------------------------------------------------
extern "C" void kernel_launch(void* const* d_in, const int* in_sizes, int n_in,
                              void* d_out, int out_size, void* d_ws, size_t ws_size,
                              hipStream_t stream) {
  const int*   index  = (const int*)d_in[0];    // [2, nnz]
  const float* value  = (const float*)d_in[1];  // [nnz]
  const float* weight = (const float*)d_in[3];  // [in_features, 64]
  const float* bias   = (const float*)d_in[4];  // [64]
  float* out = (float*)d_out;

  const int nnz = in_sizes[1];
  const int n   = out_size / 64;
  const int* rows = index;
  const int* cols = index + nnz;

  const int numScanBlocks = (n + 1023) >> 10;
  const size_t A256 = 255;
  size_t rs_off    = 0;                                                  // row_start: (n+1) u32
  size_t fill_off  = ((rs_off   + (size_t)(n + 1) * 4) + A256) & ~A256;  // counts/fill: n u32
  size_t bs_off    = ((fill_off + (size_t)n * 4)       + A256) & ~A256;  // block sums
  size_t pairs_off = ((bs_off   + (size_t)(numScanBlocks + 1) * 4) + A256) & ~A256;
  size_t needed    = pairs_off + (size_t)nnz * 8;                        // binned records

  if (ws_size >= needed) {
    unsigned* row_start = (unsigned*)((char*)d_ws + rs_off);
    unsigned* fill      = (unsigned*)((char*)d_ws + fill_off);
    unsigned* bsums     = (unsigned*)((char*)d_ws + bs_off);
    uint2*    pairs     = (uint2*)  ((char*)d_ws + pairs_off);

    k_zero      <<<(n + 255) / 256, 256, 0, stream>>>(fill, n);
    k_hist      <<<(nnz + 255) / 256, 256, 0, stream>>>(rows, fill, nnz);
    k_scan_block<<<numScanBlocks, 256, 0, stream>>>(fill, row_start, bsums, n);
    k_scan_sums <<<1, 1, 0, stream>>>(bsums, numScanBlocks);
    k_finalize  <<<(n + 1 + 255) / 256, 256, 0, stream>>>(row_start, fill, bsums, n, (unsigned)nnz);
    k_scatter   <<<(nnz + 255) / 256, 256, 0, stream>>>(rows, cols, value, fill, pairs, nnz);
    k_spmm_wmma <<<(n + 15) / 16, 32, 0, stream>>>(pairs, row_start, weight, bias, out, n);
  } else {
    k_init_out<<<(n * 64 + 255) / 256, 256, 0, stream>>>(out, bias, n * 64);
    long long tot = (long long)nnz * 8;
    k_atomic_spmm<<<(unsigned)((tot + 255) / 256), 256, 0, stream>>>(rows, cols, value, weight, out, nnz);
  }
}